// GCN_83081847374393
// MI455X (gfx1250) — compile-verified
//
#include <hip/hip_runtime.h>

typedef __attribute__((ext_vector_type(2))) float v2f;
typedef __attribute__((ext_vector_type(8))) float v8f;

#define EMB    128
#define HID    256
#define NOUT   16
#define NSRC0  50000
#define NDST0  20000
#define NWORDS 50

// ---------------- utility kernels ----------------

__global__ void k_zero4(float4* __restrict__ p4, long n4) {
  long i = (long)blockIdx.x * blockDim.x + threadIdx.x;
  if (i < n4) p4[i] = make_float4(0.f, 0.f, 0.f, 0.f);
}

// deg layout: [out0: NSRC0][in0: NDST0][out1: NDST0][in1: n_dst1]
__global__ void k_deg(const int* __restrict__ src0, const int* __restrict__ dst0, int e0,
                      const int* __restrict__ src1, const int* __restrict__ dst1, int e1,
                      float* __restrict__ deg) {
  int i = blockIdx.x * blockDim.x + threadIdx.x;
  if (i < e0) {
    atomicAdd(&deg[src0[i]], 1.0f);
    atomicAdd(&deg[NSRC0 + dst0[i]], 1.0f);
  } else if (i < e0 + e1) {
    int j = i - e0;
    atomicAdd(&deg[NSRC0 + NDST0 + src1[j]], 1.0f);
    atomicAdd(&deg[NSRC0 + 2 * NDST0 + dst1[j]], 1.0f);
  }
}

__global__ void k_rsqrt(float* __restrict__ deg, int n) {
  int i = blockIdx.x * blockDim.x + threadIdx.x;
  if (i < n) deg[i] = rsqrtf(fmaxf(deg[i], 1.0f));
}

// x[node, c] = mean_w table[user_word[node,w], c] * rsqrt(deg_out0[node])
__global__ void k_embed(const int* __restrict__ user_word,
                        const float* __restrict__ table,
                        const float* __restrict__ rdeg_out0,
                        float* __restrict__ x) {
  int node = blockIdx.x;
  int c    = threadIdx.x;
  const int* w = user_word + (long)node * NWORDS;
  float s = 0.0f;
  #pragma unroll 5
  for (int i = 0; i < NWORDS; ++i) {
    s += table[(long)w[i] * EMB + c];
  }
  x[(long)node * EMB + c] = s * (1.0f / (float)NWORDS) * rdeg_out0[node];
}

// agg0[dst, :] += x[src, :]   one wave (32 lanes) per edge, 4 channels/lane
__global__ void k_scatter0(const int* __restrict__ src, const int* __restrict__ dst,
                           const float* __restrict__ x, float* __restrict__ agg,
                           int nEdges) {
  int e = blockIdx.x * 8 + (threadIdx.x >> 5);     // 8 edges per 256-thread block
  if (e >= nEdges) return;
  int lane = threadIdx.x & 31;
  int s = src[e];
  int d = dst[e];
  const float4* xs = (const float4*)(x + (long)s * EMB);
  float*        ad = agg + (long)d * EMB + lane * 4;
  float4 v = xs[lane];                              // global_load_b128, coalesced
  atomicAdd(ad + 0, v.x);
  atomicAdd(ad + 1, v.y);
  atomicAdd(ad + 2, v.z);
  atomicAdd(ad + 3, v.w);
}

// agg1[dst, :] += h1[src, :] * rsqrt(deg_out1[src])   64 lanes/edge, 4 ch/lane
__global__ void k_scatter1(const int* __restrict__ src, const int* __restrict__ dst,
                           const float* __restrict__ h,
                           const float* __restrict__ rdeg_out1,
                           float* __restrict__ agg, int nEdges) {
  int e = blockIdx.x * 4 + (threadIdx.x >> 6);     // 4 edges per 256-thread block
  if (e >= nEdges) return;
  int t = threadIdx.x & 63;
  int s = src[e];
  int d = dst[e];
  float rs = rdeg_out1[s];
  const float4* hs = (const float4*)(h + (long)s * HID);
  float*        ad = agg + (long)d * HID + t * 4;
  float4 v = hs[t];
  atomicAdd(ad + 0, v.x * rs);
  atomicAdd(ad + 1, v.y * rs);
  atomicAdd(ad + 2, v.z * rs);
  atomicAdd(ad + 3, v.w * rs);
}

// ---------------- WMMA GEMMs (f32 16x16x4) ----------------
// h1 = relu((agg0 * rdeg_in0[:,None]) @ W1 + b1)   M=NDST0, K=EMB, N=HID
// One wave computes a 16x64 tile (4 accumulators) => 4x A reuse per load.
__global__ void k_gemm1(const float* __restrict__ agg,
                        const float* __restrict__ rdeg_in,
                        const float* __restrict__ W,
                        const float* __restrict__ bias,
                        float* __restrict__ h,
                        int totalTiles) {
  int wave = (int)((blockIdx.x * blockDim.x + threadIdx.x) >> 5);
  if (wave >= totalTiles) return;        // uniform per-wave branch
  int lane = threadIdx.x & 31;
  int l16  = lane & 15;
  int hi   = lane >> 4;

  const int nGroups = HID / 64;          // 4 groups of 64 columns
  int nTile = wave & (nGroups - 1);
  int mTile = wave / nGroups;
  int m0 = mTile * 16;
  int n0 = nTile * 64;

  int row = m0 + l16;
  float rs = rdeg_in[row];
  const float* ar = agg + (long)row * EMB;

  v8f c0 = {}, c1 = {}, c2 = {}, c3 = {};
  #pragma unroll 4
  for (int k0 = 0; k0 < EMB; k0 += 4) {
    int k = k0 + hi * 2;
    v2f a;
    a.x = ar[k]     * rs;
    a.y = ar[k + 1] * rs;
    const float* w0 = W + (long)k       * HID + n0 + l16;
    const float* w1 = W + (long)(k + 1) * HID + n0 + l16;
    v2f b0, b1, b2, b3;
    b0.x = w0[ 0]; b0.y = w1[ 0];
    b1.x = w0[16]; b1.y = w1[16];
    b2.x = w0[32]; b2.y = w1[32];
    b3.x = w0[48]; b3.y = w1[48];
    c0 = __builtin_amdgcn_wmma_f32_16x16x4_f32(false, a, false, b0, (short)0, c0, false, false);
    c1 = __builtin_amdgcn_wmma_f32_16x16x4_f32(false, a, false, b1, (short)0, c1, false, false);
    c2 = __builtin_amdgcn_wmma_f32_16x16x4_f32(false, a, false, b2, (short)0, c2, false, false);
    c3 = __builtin_amdgcn_wmma_f32_16x16x4_f32(false, a, false, b3, (short)0, c3, false, false);
  }

  #pragma unroll
  for (int j = 0; j < 4; ++j) {
    v8f c = (j == 0) ? c0 : (j == 1) ? c1 : (j == 2) ? c2 : c3;
    int col = n0 + j * 16 + l16;
    float bb = bias[col];
    #pragma unroll
    for (int v = 0; v < 8; ++v) {
      int r = m0 + v + hi * 8;
      h[(long)r * HID + col] = fmaxf(c[v] + bb, 0.0f);
    }
  }
}

// out = relu((agg1 * rdeg_in1[:,None]) @ W2 + b2)   M=n_dst1, K=HID, N=NOUT(16)
__global__ void k_gemm2(const float* __restrict__ agg,
                        const float* __restrict__ rdeg_in,
                        const float* __restrict__ W,
                        const float* __restrict__ bias,
                        float* __restrict__ out,
                        int M) {
  int wave = (int)((blockIdx.x * blockDim.x + threadIdx.x) >> 5);
  int mTiles = (M + 15) >> 4;
  if (wave >= mTiles) return;            // uniform per-wave branch
  int lane = threadIdx.x & 31;
  int l16  = lane & 15;
  int hi   = lane >> 4;

  int m0  = wave * 16;
  int row = m0 + l16;
  bool ok = row < M;
  // clamp address + zero the scale so all 32 lanes run the WMMA with EXEC all-1s
  float rs = ok ? rdeg_in[row] : 0.0f;
  const float* ar = agg + (long)(ok ? row : 0) * HID;

  v8f c = {};
  #pragma unroll 8
  for (int k0 = 0; k0 < HID; k0 += 4) {
    int k = k0 + hi * 2;
    v2f a, b;
    a.x = ar[k]     * rs;
    a.y = ar[k + 1] * rs;
    b.x = W[(long)k       * NOUT + l16];
    b.y = W[(long)(k + 1) * NOUT + l16];
    c = __builtin_amdgcn_wmma_f32_16x16x4_f32(false, a, false, b,
                                              (short)0, c, false, false);
  }

  float bb = bias[l16];
  #pragma unroll
  for (int v = 0; v < 8; ++v) {
    int r = m0 + v + hi * 8;
    if (r < M) out[(long)r * NOUT + l16] = fmaxf(c[v] + bb, 0.0f);
  }
}

__global__ void k_labels(const int* __restrict__ labels, float* __restrict__ out,
                         int n, long off) {
  int i = blockIdx.x * blockDim.x + threadIdx.x;
  if (i < n) out[off + i] = (float)labels[i];
}

// ---------------- launch ----------------
extern "C" void kernel_launch(void* const* d_in, const int* in_sizes, int n_in,
                              void* d_out, int out_size, void* d_ws, size_t ws_size,
                              hipStream_t stream) {
  const int*   user_word  = (const int*)d_in[0];
  const int*   labels     = (const int*)d_in[1];
  const int*   src0       = (const int*)d_in[2];
  const int*   dst0       = (const int*)d_in[3];
  const int*   src1       = (const int*)d_in[4];
  const int*   dst1       = (const int*)d_in[5];
  const float* word_table = (const float*)d_in[6];
  const float* W1         = (const float*)d_in[7];
  const float* b1         = (const float*)d_in[8];
  const float* W2         = (const float*)d_in[9];
  const float* b2         = (const float*)d_in[10];

  int n_dst1 = in_sizes[1];
  int e0     = in_sizes[2];
  int e1     = in_sizes[4];

  float* ws = (float*)d_ws;
  long deg_total = (long)NSRC0 + NDST0 + NDST0 + n_dst1;   // 95000
  float* deg       = ws;
  float* rdeg_out0 = deg;
  float* rdeg_in0  = deg + NSRC0;
  float* rdeg_out1 = deg + NSRC0 + NDST0;
  float* rdeg_in1  = deg + NSRC0 + 2 * NDST0;
  float* agg0 = ws + deg_total;                       // NDST0 x EMB
  float* agg1 = agg0 + (long)NDST0 * EMB;             // n_dst1 x HID
  float* x    = agg1 + (long)n_dst1 * HID;            // NSRC0 x EMB (fully written)
  float* h1   = x + (long)NSRC0 * EMB;                // NDST0 x HID (fully written)

  // zero the accumulated regions (degrees + both agg buffers); sizes are
  // multiples of 4 floats (95000 + 2.56M + n_dst1*256)
  long zeroN  = deg_total + (long)NDST0 * EMB + (long)n_dst1 * HID;
  long zeroN4 = (zeroN + 3) / 4;
  k_zero4<<<(int)((zeroN4 + 255) / 256), 256, 0, stream>>>((float4*)ws, zeroN4);

  int degThreads = e0 + e1;
  k_deg<<<(degThreads + 255) / 256, 256, 0, stream>>>(src0, dst0, e0, src1, dst1, e1, deg);

  k_rsqrt<<<(int)((deg_total + 255) / 256), 256, 0, stream>>>(deg, (int)deg_total);

  k_embed<<<NSRC0, EMB, 0, stream>>>(user_word, word_table, rdeg_out0, x);

  k_scatter0<<<(e0 + 7) / 8, 256, 0, stream>>>(src0, dst0, x, agg0, e0);

  int tiles1 = (NDST0 / 16) * (HID / 64);   // 5000 waves of 16x64 tiles
  k_gemm1<<<(tiles1 + 7) / 8, 256, 0, stream>>>(agg0, rdeg_in0, W1, b1, h1, tiles1);

  k_scatter1<<<(e1 + 3) / 4, 256, 0, stream>>>(src1, dst1, h1, rdeg_out1, agg1, e1);

  int mTiles2 = (n_dst1 + 15) / 16;
  k_gemm2<<<(mTiles2 + 7) / 8, 256, 0, stream>>>(agg1, rdeg_in1, W2, b2, (float*)d_out, n_dst1);

  long labOff = (long)n_dst1 * NOUT;
  k_labels<<<(n_dst1 + 255) / 256, 256, 0, stream>>>(labels, (float*)d_out, n_dst1, labOff);
}